// GCNBlock_66932770341394
// MI455X (gfx1250) — compile-verified
//
#include <hip/hip_runtime.h>

typedef __attribute__((ext_vector_type(2))) float        v2f;
typedef __attribute__((ext_vector_type(8))) float        v8f;
typedef __attribute__((ext_vector_type(4))) unsigned int v4u;
typedef __attribute__((ext_vector_type(8))) int          v8i;
typedef __attribute__((ext_vector_type(4))) int          v4i;

#define HIDDEN 128
#define LDA 129            // padded LDS stride (129 % 64 == 1 -> conflict-free column reads)
#define BN_EPS 1e-5f

#ifndef USE_TDM
#define USE_TDM 1
#endif

// ---------------- K1: init accumulators ----------------
__global__ void k_init(float* __restrict__ agg, float* __restrict__ dinv,
                       float* __restrict__ sums, float* __restrict__ sumsq,
                       int n_nodes, long long total) {
  long long t = (long long)blockIdx.x * blockDim.x + threadIdx.x;
  if (t < total) agg[t] = 0.0f;
  if (t < n_nodes) dinv[t] = 1.0f;              // self-loop contributes 1 to degree
  if (t < HIDDEN) { sums[t] = 0.0f; sumsq[t] = 0.0f; }
}

// ---------------- K2: degree of A+I via atomics on target ----------------
__global__ void k_degree(const int* __restrict__ dst, float* __restrict__ deg, int E) {
  int e = blockIdx.x * blockDim.x + threadIdx.x;
  if (e < E) atomicAdd(&deg[dst[e]], 1.0f);
}

// ---------------- K3: deg -> deg^{-1/2} in place ----------------
__global__ void k_dinv(float* __restrict__ deg, int N) {
  int i = blockIdx.x * blockDim.x + threadIdx.x;
  if (i < N) deg[i] = rsqrtf(deg[i]);           // deg >= 1 guaranteed
}

// ---------------- K4: XW = X @ W via V_WMMA_F32_16X16X4_F32 ----------------
// Block: 256 threads = 8 waves. Block computes a 16(M) x 128(N) tile.
// A tile staged Global->LDS by the Tensor Data Mover (wave 0 issues one
// tensor_load_to_lds; TDM pad_enable inserts 1 DWORD per 128 -> LDS stride 129).
__global__ __launch_bounds__(256) void k_gemm_wmma(const float* __restrict__ X,
                                                   const float* __restrict__ W,
                                                   float* __restrict__ XW, int N) {
  __shared__ float As[16 * LDA];
  const int row0 = blockIdx.x * 16;
  const int tid  = threadIdx.x;
  const int wave = tid >> 5;
  const int lane = tid & 31;

#if USE_TDM && __has_builtin(__builtin_amdgcn_tensor_load_to_lds)
  if (wave == 0) {                       // one TDM op per block; EXEC all-1s in wave 0
    unsigned long long ga = (unsigned long long)(X + (size_t)row0 * HIDDEN);
    unsigned int ldsa = (unsigned int)(unsigned long long)As;   // generic low 32b = LDS byte addr
    int rows = N - row0; if (rows > 16) rows = 16;              // OOB rows read as zero
    v4u g0;
    g0.x = 1u;                                                  // count=1 user descriptor
    g0.y = ldsa;                                                // lds_addr
    g0.z = (unsigned int)ga;                                    // global_addr[31:0]
    g0.w = (unsigned int)((ga >> 32) & 0x01FFFFFFull) | (2u << 30);  // global_addr[56:32] | type=2
    v8i g1;
    g1[0] = (int)((2u << 16) | (1u << 20) | (6u << 22));  // data_size=4B | pad_enable | interval=128 DW (amount=1 DW)
    g1[1] = (int)((unsigned)HIDDEN << 16);                // tensor_dim0 = 128
    g1[2] = (int)((unsigned)rows << 16);                  // tensor_dim1 = valid rows
    g1[3] = (int)((unsigned)HIDDEN << 16);                // tile_dim0 = 128
    g1[4] = 16;                                           // tile_dim1 = 16 (tile_dim2 = 0)
    g1[5] = HIDDEN;                                       // tensor_dim0_stride = 128
    g1[6] = 0;                                            // stride0[47:32]=0, stride1[15:0]=0
    g1[7] = 0;
    v4i zz4 = {0, 0, 0, 0};                               // groups 2/3 unused (<=2D tensor)
    v8i zz8 = {0, 0, 0, 0, 0, 0, 0, 0};
    __builtin_amdgcn_tensor_load_to_lds(g0, g1, zz4, zz4, zz8, 0);
    __builtin_amdgcn_s_wait_tensorcnt(0);
  }
#else
  for (int i = tid; i < 16 * HIDDEN; i += 256) {
    int m = i >> 7, k = i & 127;
    int r = row0 + m;
    As[m * LDA + k] = (r < N) ? X[(long long)r * HIDDEN + k] : 0.0f;
  }
#endif
  __syncthreads();

  const int n0   = wave * 16;
  const int half = lane >> 4;    // 0: K pair {0,1}; 1: K pair {2,3}
  const int ml   = lane & 15;    // M for A-frag, N for B-frag

  v8f c = {};
#pragma unroll
  for (int k = 0; k < HIDDEN; k += 4) {
    // A 16x4 f32 frag: VGPR0 = K=0 (lanes 0-15) / K=2 (lanes 16-31); VGPR1 = K=1 / K=3
    v2f a, b;
    a.x = As[ml * LDA + k + 2 * half];
    a.y = As[ml * LDA + k + 2 * half + 1];
    // B 4x16 f32 frag: row (K) striped across VGPRs/halves, N across lanes
    b.x = W[(k + 2 * half) * HIDDEN + n0 + ml];
    b.y = W[(k + 2 * half + 1) * HIDDEN + n0 + ml];
    c = __builtin_amdgcn_wmma_f32_16x16x4_f32(false, a, false, b, (short)0, c, false, false);
  }

  // C/D 16x16 f32: VGPR r -> M = r (lanes 0-15) / M = r+8 (lanes 16-31); N = lane&15
#pragma unroll
  for (int r = 0; r < 8; ++r) {
    int row = row0 + r + 8 * half;
    if (row < N) XW[(long long)row * HIDDEN + n0 + ml] = c[r];
  }
}

// ---------------- K5: edge gather-scale-scatter (one wave per edge) ----------------
// Edge index / norm are wave-uniform: force scalar (SMEM) loads via readfirstlane.
__global__ __launch_bounds__(256) void k_scatter(const int* __restrict__ src,
                                                 const int* __restrict__ dst,
                                                 const float* __restrict__ XW,
                                                 const float* __restrict__ dinv,
                                                 float* __restrict__ agg, int E) {
  const int lane = threadIdx.x & 31;
  int e = blockIdx.x * 8 + (threadIdx.x >> 5);   // 8 waves per 256-thread block
  if (e >= E) return;
  int r    = __builtin_amdgcn_readfirstlane(src[e]);
  int cidx = __builtin_amdgcn_readfirstlane(dst[e]);
  float nrm = dinv[r] * dinv[cidx];              // uniform scalar addresses
  const float4 v = *(const float4*)(XW + (long long)r * HIDDEN + lane * 4);
  float* out = agg + (long long)cidx * HIDDEN + lane * 4;
  atomicAdd(out + 0, v.x * nrm);
  atomicAdd(out + 1, v.y * nrm);
  atomicAdd(out + 2, v.z * nrm);
  atomicAdd(out + 3, v.w * nrm);
}

// ---------------- K6: add self-loop + bias; accumulate per-feature stats ----------------
// blockDim = 128 (thread = feature), each block reduces 64 nodes -> 2 atomics/thread
__global__ __launch_bounds__(128) void k_finalize(float* __restrict__ agg,
                                                  const float* __restrict__ XW,
                                                  const float* __restrict__ dinv,
                                                  const float* __restrict__ bias,
                                                  float* __restrict__ sums,
                                                  float* __restrict__ sumsq, int N) {
  const int f  = threadIdx.x;
  const int i0 = blockIdx.x * 64;
  const float bf = bias[f];
  float s = 0.0f, sq = 0.0f;
  for (int j = 0; j < 64; ++j) {
    int i = i0 + j;
    if (i >= N) break;
    float di = dinv[i];
    long long idx = (long long)i * HIDDEN + f;
    float v = agg[idx] + di * di * XW[idx] + bf;
    agg[idx] = v;
    s += v; sq += v * v;
  }
  atomicAdd(&sums[f], s);
  atomicAdd(&sumsq[f], sq);
}

// ---------------- K7: mean / inv-std ----------------
__global__ void k_stats(const float* __restrict__ sums, const float* __restrict__ sumsq,
                        float* __restrict__ mean, float* __restrict__ inv, int N) {
  int f = threadIdx.x;
  float m = sums[f] / (float)N;
  float v = sumsq[f] / (float)N - m * m;       // biased variance
  mean[f] = m;
  inv[f]  = rsqrtf(v + BN_EPS);
}

// ---------------- K8: BN affine + ReLU ----------------
__global__ void k_bnrelu(float* __restrict__ out, const float* __restrict__ mean,
                         const float* __restrict__ inv, const float* __restrict__ gamma,
                         const float* __restrict__ beta, long long total) {
  long long t = (long long)blockIdx.x * blockDim.x + threadIdx.x;
  if (t >= total) return;
  int f = (int)(t & (HIDDEN - 1));
  float h = (out[t] - mean[f]) * inv[f] * gamma[f] + beta[f];
  out[t] = fmaxf(h, 0.0f);
}

extern "C" void kernel_launch(void* const* d_in, const int* in_sizes, int n_in,
                              void* d_out, int out_size, void* d_ws, size_t ws_size,
                              hipStream_t stream) {
  const float* x     = (const float*)d_in[0];
  const int*   ei    = (const int*)d_in[1];     // [2, E] int32
  const float* W     = (const float*)d_in[2];
  const float* b     = (const float*)d_in[3];
  const float* gamma = (const float*)d_in[4];
  const float* beta  = (const float*)d_in[5];

  const int N = in_sizes[0] / HIDDEN;
  const int E = in_sizes[1] / 2;
  const int* src = ei;        // edge_index[0] (source / row)
  const int* dst = ei + E;    // edge_index[1] (target / col)

  float* agg = (float*)d_out;

  char* ws = (char*)d_ws;
  float* xw    = (float*)ws;  ws += (size_t)N * HIDDEN * sizeof(float);
  float* dinv  = (float*)ws;  ws += (size_t)N * sizeof(float);
  float* sums  = (float*)ws;  ws += HIDDEN * sizeof(float);
  float* sumsq = (float*)ws;  ws += HIDDEN * sizeof(float);
  float* mean  = (float*)ws;  ws += HIDDEN * sizeof(float);
  float* inv   = (float*)ws;

  const long long total = (long long)N * HIDDEN;

  k_init<<<(int)((total + 255) / 256), 256, 0, stream>>>(agg, dinv, sums, sumsq, N, total);
  k_degree<<<(E + 255) / 256, 256, 0, stream>>>(dst, dinv, E);
  k_dinv<<<(N + 255) / 256, 256, 0, stream>>>(dinv, N);
  k_gemm_wmma<<<(N + 15) / 16, 256, 0, stream>>>(x, W, xw, N);
  k_scatter<<<(E + 7) / 8, 256, 0, stream>>>(src, dst, xw, dinv, agg, E);
  k_finalize<<<(N + 63) / 64, 128, 0, stream>>>(agg, xw, dinv, b, sums, sumsq, N);
  k_stats<<<1, 128, 0, stream>>>(sums, sumsq, mean, inv, N);
  k_bnrelu<<<(int)((total + 255) / 256), 256, 0, stream>>>(agg, mean, inv, gamma, beta, total);
}